// CannyEdge_764504179418
// MI455X (gfx1250) — compile-verified
//
#include <hip/hip_runtime.h>
#include <math.h>

typedef __attribute__((ext_vector_type(2))) float v2f;
typedef __attribute__((ext_vector_type(8))) float v8f;

#define IMG_H 1024
#define IMG_W 1024
#define BATCH 8
#define NPIX (BATCH * IMG_H * IMG_W)
#define TILES_PER_IMG (64 * 64)

// select w[d] for d in [0,4], lane-divergent d (cndmask chain, no scratch)
__device__ __forceinline__ float wsel(int d, float w0, float w1, float w2, float w3, float w4) {
    float r = 0.0f;
    r = (d == 0) ? w0 : r;
    r = (d == 1) ? w1 : r;
    r = (d == 2) ? w2 : r;
    r = (d == 3) ? w3 : r;
    r = (d == 4) ? w4 : r;
    return r;
}

// ---------------------------------------------------------------------------
// 1) Gaussian blur: one wave per 16x16 output tile.
//    Horizontal 1D pass in LDS (VALU), vertical 1D pass = 5x V_WMMA_F32_16X16X4_F32
//    D(16x16) = sum_c A_c(16x4 banded weights) x B_c(4x16 rows of h-blurred tile)
// ---------------------------------------------------------------------------
__global__ __launch_bounds__(256) void k_blur_wmma(const float* __restrict__ img,
                                                   const float* __restrict__ gk,
                                                   float* __restrict__ out) {
    __shared__ float lds[8 * 740]; // per wave: 20x21 patch (420) + 20x16 htmp (320)
    const int lane = threadIdx.x & 31;
    const int wave = threadIdx.x >> 5;
    float* patch = &lds[wave * 740];
    float* htmp  = patch + 420;

    const int tile = blockIdx.x * 8 + wave;
    const int b  = tile >> 12;         // / 4096
    const int t2 = tile & 4095;
    const int ty = t2 >> 6;
    const int tx = t2 & 63;
    const float* base = img + (size_t)b * (IMG_H * IMG_W);

    // separable 1D weights = column sums of the provided (separable) 5x5 kernel
    float w0 = 0.f, w1 = 0.f, w2 = 0.f, w3 = 0.f, w4 = 0.f;
#pragma unroll
    for (int i = 0; i < 5; ++i) {
        w0 += gk[i * 5 + 0]; w1 += gk[i * 5 + 1]; w2 += gk[i * 5 + 2];
        w3 += gk[i * 5 + 3]; w4 += gk[i * 5 + 4];
    }

    // load 20x20 patch with reflect padding (np.pad 'reflect', no edge dup)
    for (int i = lane; i < 400; i += 32) {
        int r = i / 20, c = i % 20;
        int gy = ty * 16 + r - 2;
        int gx = tx * 16 + c - 2;
        gy = (gy < 0) ? -gy : ((gy >= IMG_H) ? (2 * IMG_H - 2 - gy) : gy);
        gx = (gx < 0) ? -gx : ((gx >= IMG_W) ? (2 * IMG_W - 2 - gx) : gx);
        patch[r * 21 + c] = base[(size_t)gy * IMG_W + gx];
    }
    __syncthreads();

    // horizontal blur: 20 rows x 16 cols into htmp
    for (int i = lane; i < 320; i += 32) {
        int r = i >> 4, n = i & 15;
        const float* p = &patch[r * 21 + n];
        htmp[i] = p[0] * w0 + p[1] * w1 + p[2] * w2 + p[3] * w3 + p[4] * w4;
    }
    __syncthreads();

    // vertical blur: D = Wv(16x20) x H(20x16), K split into 5 chunks of 4
    const int hi = lane >> 4;   // half-wave select
    const int mn = lane & 15;   // M for A, N for B/D
    v8f acc = {};
#pragma unroll
    for (int c4 = 0; c4 < 20; c4 += 4) {
        const int k0 = c4 + hi * 2;
        // A[m][k] = w[k-m] when 0 <= k-m <= 4 (banded upper-triangular-ish)
        v2f a, bm;
        a.x = wsel(k0 - mn,     w0, w1, w2, w3, w4);
        a.y = wsel(k0 + 1 - mn, w0, w1, w2, w3, w4);
        // B[k][n] = htmp[k][n]
        bm.x = htmp[k0 * 16 + mn];
        bm.y = htmp[(k0 + 1) * 16 + mn];
        acc = __builtin_amdgcn_wmma_f32_16x16x4_f32(false, a, false, bm,
                                                    (short)0, acc, false, false);
    }

    // C/D layout: VGPR r -> row r (lanes 0-15) / row r+8 (lanes 16-31), col = lane&15
    float* obase = out + (size_t)b * (IMG_H * IMG_W) + (size_t)(ty * 16) * IMG_W + tx * 16;
#pragma unroll
    for (int r = 0; r < 8; ++r)
        obase[(size_t)(r + hi * 8) * IMG_W + mn] = acc[r];
}

// ---------------------------------------------------------------------------
// 2) Sobel (zero pad 1, cross-correlation as XLA conv) -> mag, theta
// ---------------------------------------------------------------------------
__global__ __launch_bounds__(256) void k_sobel(const float* __restrict__ blur,
                                               const float* __restrict__ sf,
                                               float* __restrict__ mag,
                                               float* __restrict__ theta) {
    int idx = blockIdx.x * 256 + threadIdx.x;
    if (idx >= NPIX) return;
    int x = idx % IMG_W;
    int t = idx / IMG_W;
    int y = t % IMG_H;
    int b = t / IMG_H;
    const float* pb = blur + (size_t)b * IMG_H * IMG_W;
    float g0 = 0.f, g1 = 0.f;
#pragma unroll
    for (int i = 0; i < 3; ++i)
#pragma unroll
        for (int j = 0; j < 3; ++j) {
            int yy = y + i - 1, xx = x + j - 1;
            float v = (yy >= 0 && yy < IMG_H && xx >= 0 && xx < IMG_W)
                          ? pb[(size_t)yy * IMG_W + xx] : 0.0f;
            g0 += v * sf[(i * 3 + j) * 2 + 0];
            g1 += v * sf[(i * 3 + j) * 2 + 1];
        }
    float th = atan2f(g0, g1) * 57.29577951308232f + 90.0f;
    th = th - 180.0f * floorf(th * (1.0f / 180.0f));   // jnp.mod(a, 180)
    mag[idx]   = sqrtf(g0 * g0 + g1 * g1);
    theta[idx] = th;
}

// ---------------------------------------------------------------------------
// 3) angle-wise NMS + double threshold -> s in {0,1,2} (weak=1, sure=2)
// ---------------------------------------------------------------------------
__device__ __forceinline__ int angle_bucket(float t) {
    if (t >= 157.5f || t < 22.5f) return 0;   // 0 deg
    if (t < 67.5f)  return 1;                 // 45 deg
    if (t < 112.5f) return 2;                 // 90 deg
    return 3;                                 // 135 deg
}

__global__ __launch_bounds__(256) void k_nms(const float* __restrict__ mag,
                                             const float* __restrict__ theta,
                                             float* __restrict__ sfield) {
    int idx = blockIdx.x * 256 + threadIdx.x;
    if (idx >= NPIX) return;
    int x = idx % IMG_W;
    int t = idx / IMG_W;
    int y = t % IMG_H;

    float m  = mag[idx];
    int   bk = angle_bucket(theta[idx]);

    // two non-center offsets of each bucket's dilation kernel
    const int dy0[4] = { 0, -1, -1, -1 };
    const int dx0[4] = { -1, 1,  0, -1 };
    const int dy1[4] = { 0,  1,  1,  1 };
    const int dx1[4] = { 1, -1,  0,  1 };

    float mx = m;
    {
        int yy = y + dy0[bk], xx = x + dx0[bk];
        if (yy >= 0 && yy < IMG_H && xx >= 0 && xx < IMG_W) {
            int n = idx + dy0[bk] * IMG_W + dx0[bk];
            if (angle_bucket(theta[n]) == bk) mx = fmaxf(mx, mag[n]);
        }
    }
    {
        int yy = y + dy1[bk], xx = x + dx1[bk];
        if (yy >= 0 && yy < IMG_H && xx >= 0 && xx < IMG_W) {
            int n = idx + dy1[bk] * IMG_W + dx1[bk];
            if (angle_bucket(theta[n]) == bk) mx = fmaxf(mx, mag[n]);
        }
    }
    float edge = (mx == m) ? m : 0.0f;
    sfield[idx] = (edge >= 100.0f) ? 2.0f : ((edge >= 50.0f) ? 1.0f : 0.0f);
}

// ---------------------------------------------------------------------------
// 4) initial connectivity: cs=dil5(sure), cw=dil5(weak); conn = cs*weak+cw*sure >= 1
// ---------------------------------------------------------------------------
__global__ __launch_bounds__(256) void k_conn0(const float* __restrict__ sfield,
                                               float* __restrict__ conn) {
    int idx = blockIdx.x * 256 + threadIdx.x;
    if (idx >= NPIX) return;
    int x = idx % IMG_W;
    int t = idx / IMG_W;
    int y = t % IMG_H;

    float s = sfield[idx];
    float cs = 0.f, cw = 0.f;
#pragma unroll
    for (int dy = -2; dy <= 2; ++dy) {
        int yy = y + dy;
        if (yy < 0 || yy >= IMG_H) continue;
#pragma unroll
        for (int dx = -2; dx <= 2; ++dx) {
            int xx = x + dx;
            if (xx < 0 || xx >= IMG_W) continue;
            float sn = sfield[idx + dy * IMG_W + dx];
            cs = fmaxf(cs, (sn >= 1.5f) ? 1.0f : 0.0f);
            cw = fmaxf(cw, (sn >= 0.5f && sn < 1.5f) ? 1.0f : 0.0f);
        }
    }
    float weak = (s >= 0.5f && s < 1.5f) ? 1.0f : 0.0f;
    float sure = (s >= 1.5f) ? 1.0f : 0.0f;
    conn[idx] = (cs * weak + cw * sure >= 1.0f) ? 1.0f : 0.0f;
}

// ---------------------------------------------------------------------------
// 5) separable 5x5 dilation, float4-vectorized (W % 4 == 0, rows never split)
// ---------------------------------------------------------------------------
__global__ __launch_bounds__(256) void k_rowmax4(const float* __restrict__ conn,
                                                 float* __restrict__ tmp) {
    int idx4 = blockIdx.x * 256 + threadIdx.x;
    if (idx4 >= NPIX / 4) return;
    int xi = idx4 & 255;                   // W/4 = 256
    size_t row = (size_t)(idx4 >> 8) * IMG_W;
    const float4* cp = reinterpret_cast<const float4*>(conn + row);
    __builtin_prefetch((const void*)(conn + row + IMG_W + (size_t)xi * 4), 0, 0);

    float4 c = cp[xi];
    float4 p = (xi > 0)   ? cp[xi - 1] : make_float4(0.f, 0.f, 0.f, 0.f);
    float4 n = (xi < 255) ? cp[xi + 1] : make_float4(0.f, 0.f, 0.f, 0.f);

    float4 r;
    r.x = fmaxf(fmaxf(fmaxf(p.z, p.w), fmaxf(c.x, c.y)), c.z);
    r.y = fmaxf(fmaxf(fmaxf(p.w, c.x), fmaxf(c.y, c.z)), c.w);
    r.z = fmaxf(fmaxf(fmaxf(c.x, c.y), fmaxf(c.z, c.w)), n.x);
    r.w = fmaxf(fmaxf(fmaxf(c.y, c.z), fmaxf(c.w, n.x)), n.y);
    reinterpret_cast<float4*>(tmp + row)[xi] = r;
}

__global__ __launch_bounds__(256) void k_colmax4(const float* __restrict__ tmp,
                                                 const float* __restrict__ sfield,
                                                 float* __restrict__ conn) {
    int idx4 = blockIdx.x * 256 + threadIdx.x;
    if (idx4 >= NPIX / 4) return;
    int xi = idx4 & 255;
    int r  = idx4 >> 8;                    // global row index across batch
    int y  = r & (IMG_H - 1);
    float4 m = make_float4(0.f, 0.f, 0.f, 0.f);
#pragma unroll
    for (int dy = -2; dy <= 2; ++dy) {
        int yy = y + dy;
        if (yy < 0 || yy >= IMG_H) continue;
        float4 v = reinterpret_cast<const float4*>(tmp + (size_t)(r + dy) * IMG_W)[xi];
        m.x = fmaxf(m.x, v.x); m.y = fmaxf(m.y, v.y);
        m.z = fmaxf(m.z, v.z); m.w = fmaxf(m.w, v.w);
    }
    float4 s = reinterpret_cast<const float4*>(sfield + (size_t)r * IMG_W)[xi];
    float4 o;
    o.x = (m.x >= 0.5f && s.x >= 0.5f) ? 1.0f : 0.0f;
    o.y = (m.y >= 0.5f && s.y >= 0.5f) ? 1.0f : 0.0f;
    o.z = (m.z >= 0.5f && s.z >= 0.5f) ? 1.0f : 0.0f;
    o.w = (m.w >= 0.5f && s.w >= 0.5f) ? 1.0f : 0.0f;
    reinterpret_cast<float4*>(conn + (size_t)r * IMG_W)[xi] = o;
}

// ---------------------------------------------------------------------------
// 6) final: 255 where connected | sure
// ---------------------------------------------------------------------------
__global__ __launch_bounds__(256) void k_final4(const float* __restrict__ conn,
                                                float* __restrict__ sfield_io) {
    int idx4 = blockIdx.x * 256 + threadIdx.x;
    if (idx4 >= NPIX / 4) return;
    float4 c = reinterpret_cast<const float4*>(conn)[idx4];
    float4 s = reinterpret_cast<const float4*>(sfield_io)[idx4];
    float4 o;
    o.x = (c.x + ((s.x >= 1.5f) ? 1.0f : 0.0f) >= 1.0f) ? 255.0f : 0.0f;
    o.y = (c.y + ((s.y >= 1.5f) ? 1.0f : 0.0f) >= 1.0f) ? 255.0f : 0.0f;
    o.z = (c.z + ((s.z >= 1.5f) ? 1.0f : 0.0f) >= 1.0f) ? 255.0f : 0.0f;
    o.w = (c.w + ((s.w >= 1.5f) ? 1.0f : 0.0f) >= 1.0f) ? 255.0f : 0.0f;
    reinterpret_cast<float4*>(sfield_io)[idx4] = o;
}

extern "C" void kernel_launch(void* const* d_in, const int* in_sizes, int n_in,
                              void* d_out, int out_size, void* d_ws, size_t ws_size,
                              hipStream_t stream) {
    const float* img = (const float*)d_in[0];   // 8x1024x1024x1
    const float* gk  = (const float*)d_in[1];   // 5x5x1x1
    const float* sf  = (const float*)d_in[2];   // 3x3x1x2
    float* out = (float*)d_out;                 // reused: blur -> s-field -> final
    float* ws0 = (float*)d_ws;                  // mag -> conn
    float* ws1 = ws0 + (size_t)NPIX;            // theta -> rowmax tmp

    const int pgrid  = (NPIX + 255) / 256;          // scalar kernels
    const int vgrid  = (NPIX / 4 + 255) / 256;      // float4 kernels
    const int tgrid  = (BATCH * TILES_PER_IMG) / 8; // 8 wave-tiles per block

    k_blur_wmma<<<tgrid, 256, 0, stream>>>(img, gk, out);          // blur -> out
    k_sobel<<<pgrid, 256, 0, stream>>>(out, sf, ws0, ws1);         // mag, theta
    k_nms<<<pgrid, 256, 0, stream>>>(ws0, ws1, out);               // s -> out
    k_conn0<<<pgrid, 256, 0, stream>>>(out, ws0);                  // conn -> ws0

    // fixed 19 hysteresis iterations (monotone dilation => same fixed point
    // as the reference's early-exit loop, and deterministic for graph replay)
    for (int it = 0; it < 19; ++it) {
        k_rowmax4<<<vgrid, 256, 0, stream>>>(ws0, ws1);
        k_colmax4<<<vgrid, 256, 0, stream>>>(ws1, out, ws0);
    }
    k_final4<<<vgrid, 256, 0, stream>>>(ws0, out);
}